// GINEConv_9826885173951
// MI455X (gfx1250) — compile-verified
//
#include <hip/hip_runtime.h>
#include <hip/hip_bf16.h>
#include <stdint.h>

#define D_FEAT 32
#define EPB    128            // edges staged per block
#define BLOCK  256            // 8 waves (wave32)
#define VEC_PER_CHUNK (EPB * D_FEAT / 4)   // 1024 float4 transfers per chunk

// ---- CDNA5 async global->LDS copy (16B per lane), non-temporal hint ----
__device__ __forceinline__ void async_g2l_b128(unsigned lds_byte_addr,
                                               const void* sbase,
                                               unsigned voffset_bytes) {
    // VFLAT GVS form: address is the 64-bit SGPR base plus the 32-bit VGPR byte offset plus the immediate
    asm volatile("global_load_async_to_lds_b128 %0, %1, %2 th:TH_LOAD_NT"
                 :
                 : "v"(lds_byte_addr), "v"(voffset_bytes), "s"(sbase)
                 : "memory");
}

__device__ __forceinline__ void wait_async_zero() {
    asm volatile("s_wait_asynccnt 0" ::: "memory");
}

__device__ __forceinline__ unsigned lds_addr_of(const void* p) {
    // generic (LDS-backed) pointer -> 32-bit LDS byte address
    return (unsigned)(size_t)(__attribute__((address_space(3))) const void*)p;
}

// ---- Kernel: per-edge relu(h_src + h_edge) scatter-add via L2 atomics ----
// (defined FIRST so the disasm snippet shows this kernel)
__global__ void __launch_bounds__(BLOCK)
gine_edge_scatter(const float* __restrict__ node_feat,
                  const float* __restrict__ edge_feat,
                  const long long* __restrict__ src,
                  const long long* __restrict__ dst,
                  float* __restrict__ out, int n_edges) {
    __shared__ __align__(16) float s_edge[EPB * D_FEAT];   // 16 KB

    const int e0 = blockIdx.x * EPB;

    // Stage this block's edge_feat chunk into LDS with async b128 copies.
    const float* gbase = edge_feat + (size_t)e0 * D_FEAT;
    for (int i = threadIdx.x; i < VEC_PER_CHUNK; i += BLOCK) {
        int e = e0 + (i >> 3);                // 8 float4 per edge
        if (e < n_edges) {
            unsigned lds = lds_addr_of(&s_edge[i * 4]);
            async_g2l_b128(lds, gbase, (unsigned)(i * 16));
        }
    }
    wait_async_zero();     // this wave's async transfers complete
    __syncthreads();       // all waves' LDS writes visible

    const int wave = threadIdx.x >> 5;
    const int lane = threadIdx.x & 31;

    // wave-per-edge, lane-per-feature
    for (int le = wave; le < EPB; le += (BLOCK / 32)) {
        int e = e0 + le;
        if (e >= n_edges) break;
        // indices are streamed once -> non-temporal, keep L2 for node/out
        long long s = __builtin_nontemporal_load(&src[e]);
        long long d = __builtin_nontemporal_load(&dst[e]);
        float m = node_feat[(size_t)s * D_FEAT + lane]   // L2-resident gather
                + s_edge[le * D_FEAT + lane];
        m = fmaxf(m, 0.0f);
        // hardware global_atomic_add_f32, resolved in L2 (target is 6.4 MB)
        unsafeAtomicAdd(&out[(size_t)d * D_FEAT + lane], m);
    }
}

// ---- Kernel: out = (1 + eps) * node_feat (vectorized float4) ----
__global__ void gine_init_out(const float4* __restrict__ nf,
                              const float* __restrict__ eps,
                              float4* __restrict__ out, int n4) {
    int i = blockIdx.x * blockDim.x + threadIdx.x;
    if (i < n4) {
        float s = 1.0f + eps[0];
        float4 v = nf[i];
        v.x *= s; v.y *= s; v.z *= s; v.w *= s;
        out[i] = v;
    }
}

extern "C" void kernel_launch(void* const* d_in, const int* in_sizes, int n_in,
                              void* d_out, int out_size, void* d_ws, size_t ws_size,
                              hipStream_t stream) {
    const float*      node_feat = (const float*)d_in[0];      // [N, 32]
    const float*      edge_feat = (const float*)d_in[1];      // [E, 32]
    const float*      eps       = (const float*)d_in[2];      // [1]
    const long long*  src       = (const long long*)d_in[3];  // [E] int64
    const long long*  dst       = (const long long*)d_in[4];  // [E] int64
    float*            out       = (float*)d_out;              // [N, 32]

    const int n_edges = in_sizes[3];
    const int n_node_elems = in_sizes[0];        // N * 32
    const int n4 = n_node_elems / 4;

    // 1) out = (1+eps) * node_feat
    gine_init_out<<<(n4 + BLOCK - 1) / BLOCK, BLOCK, 0, stream>>>(
        (const float4*)node_feat, eps, (float4*)out, n4);

    // 2) scatter-add relu(h_src + h_edge)
    const int nblocks = (n_edges + EPB - 1) / EPB;
    gine_edge_scatter<<<nblocks, BLOCK, 0, stream>>>(
        node_feat, edge_feat, src, dst, out, n_edges);
}